// RNNTLoss_78391743087293
// MI455X (gfx1250) — compile-verified
//
#include <hip/hip_runtime.h>
#include <hip/hip_bf16.h>
#include <math.h>
#include <stdint.h>

constexpr int Bc = 4;
constexpr int Tc = 256;
constexpr int Uc = 100;
constexpr int U1 = 101;     // U + 1
constexpr int Vc = 1024;
constexpr int ROWS = Bc * Tc * U1;   // 103424 softmax rows

// ---------------------------------------------------------------------------
// Issue one 4KB row (1024 f32) as 8 async global->LDS B128 copies.
// Each instruction moves 32 lanes x 16B = 512B. Tracked by ASYNCcnt.
// ---------------------------------------------------------------------------
__device__ __forceinline__ void issue_row_async(const float* gsrc,
                                                unsigned ldsbase, int lane) {
  unsigned long long g = (unsigned long long)(uintptr_t)gsrc + (unsigned)(lane * 16);
  unsigned l = ldsbase + (unsigned)(lane * 16);
#pragma unroll
  for (int k = 0; k < 8; ++k) {
    asm volatile("global_load_async_to_lds_b128 %0, %1, off"
                 :: "v"(l), "v"(g) : "memory");
    g += 512;
    l += 512;
  }
}

// ---------------------------------------------------------------------------
// Kernel 1: per-row logsumexp over V=1024; emit lp_blank and gathered lp_emit.
// One wave32 per row, async-LDS double buffering across the grid-stride loop.
// LDS: 8 waves * 2 buffers * 4KB = 64KB per block.
// ---------------------------------------------------------------------------
__global__ __launch_bounds__(256) void rnnt_lse_kernel(
    const float* __restrict__ logits, const int* __restrict__ targets,
    float* __restrict__ lpb, float* __restrict__ lpe) {
  __shared__ float ldsbuf[8 * 2 * 1024];
  const int warp = threadIdx.x >> 5;
  const int lane = threadIdx.x & 31;
  const int nwaves = gridDim.x * (blockDim.x >> 5);
  int row = blockIdx.x * (blockDim.x >> 5) + warp;
  if (row >= ROWS) return;

  // low 32 bits of a flat shared-memory address == LDS byte offset
  const unsigned base0 =
      (unsigned)(unsigned long long)(uintptr_t)(const void*)&ldsbuf[warp * 2048];

  int buf = 0;
  issue_row_async(logits + (size_t)row * Vc, base0, lane);

  for (; row < ROWS; row += nwaves) {
    const int nxt = row + nwaves;
    const bool hn = nxt < ROWS;
    if (hn) issue_row_async(logits + (size_t)nxt * Vc,
                            base0 + (unsigned)((buf ^ 1) * 4096), lane);
    // async loads retire in order: <=8 outstanding means current row landed
    if (hn) { asm volatile("s_wait_asynccnt 0x8" ::: "memory"); }
    else    { asm volatile("s_wait_asynccnt 0x0" ::: "memory"); }

    const float* rb = &ldsbuf[warp * 2048 + buf * 1024];
    const float4* vp = (const float4*)rb;
    float4 q[8];
#pragma unroll
    for (int k = 0; k < 8; ++k) q[k] = vp[lane + (k << 5)];  // ds_load_b128 x8

    float m = -INFINITY;
#pragma unroll
    for (int k = 0; k < 8; ++k)
      m = fmaxf(m, fmaxf(fmaxf(q[k].x, q[k].y), fmaxf(q[k].z, q[k].w)));
#pragma unroll
    for (int off = 16; off > 0; off >>= 1) m = fmaxf(m, __shfl_xor(m, off));

    float s = 0.f;
#pragma unroll
    for (int k = 0; k < 8; ++k)
      s += __expf(q[k].x - m) + __expf(q[k].y - m) +
           __expf(q[k].z - m) + __expf(q[k].w - m);
#pragma unroll
    for (int off = 16; off > 0; off >>= 1) s += __shfl_xor(s, off);

    const float lse = m + __logf(s);

    if (lane == 0) {
      const int u  = row % U1;
      const int bt = row / U1;            // b*T + t
      lpb[row] = q[0].x - lse;            // lane0/k0 holds elements 0..3
      if (u < Uc) {
        const int b = bt / Tc;
        int tgt = targets[b * Uc + u];
        tgt = tgt < 0 ? 0 : (tgt > Vc - 1 ? Vc - 1 : tgt);
        lpe[bt * Uc + u] = rb[tgt] - lse;
      }
    }
    buf ^= 1;
  }
}

// ---------------------------------------------------------------------------
// Kernel 2: alpha recursion. One wave32 per batch; lane owns u = 4*lane..+3.
// Exclusive add-scan (Ecum) + inclusive logaddexp-scan via wave shuffles.
// ---------------------------------------------------------------------------
__device__ __forceinline__ float laef(float a, float b) {
  if (a == -INFINITY) return b;
  if (b == -INFINITY) return a;
  const float m = fmaxf(a, b);
  return m + log1pf(__expf(-fabsf(a - b)));
}

__global__ __launch_bounds__(32) void rnnt_dp_kernel(
    const float* __restrict__ lpb, const float* __restrict__ lpe,
    const int* __restrict__ loglen, const int* __restrict__ tlen,
    float* __restrict__ out) {
  const int b = blockIdx.x;
  const int lane = threadIdx.x & 31;
  const int tl = loglen[b] - 1;
  const int ul = tlen[b];
  const int u0 = lane * 4;

  bool val[4];
#pragma unroll
  for (int k = 0; k < 4; ++k) val[k] = (u0 + k) < U1;

  float alpha[4];

  for (int t = 0; t < Tc; ++t) {
    // --- exclusive prefix sum of emit probs (e[U]=0 pad) -> Ecum ---
    float e[4];
#pragma unroll
    for (int k = 0; k < 4; ++k) {
      const int u = u0 + k;
      e[k] = (u < Uc) ? lpe[(b * Tc + t) * Uc + u] : 0.f;
    }
    float inc[4];
    inc[0] = e[0];
    inc[1] = inc[0] + e[1];
    inc[2] = inc[1] + e[2];
    inc[3] = inc[2] + e[3];
    float tot = inc[3];
#pragma unroll
    for (int off = 1; off < 32; off <<= 1) {
      float n = __shfl_up(tot, off);
      if (lane >= off) tot += n;
    }
    float ex = __shfl_up(tot, 1);
    if (lane == 0) ex = 0.f;
    float E[4];
#pragma unroll
    for (int k = 0; k < 4; ++k) E[k] = ex + inc[k] - e[k];

    if (t == 0) {
#pragma unroll
      for (int k = 0; k < 4; ++k) alpha[k] = val[k] ? E[k] : -INFINITY;
    } else {
      // a = alpha_{t-1} + lp_blank[t-1]; alpha_t = E + cumlogsumexp(a - E)
      float z[4];
#pragma unroll
      for (int k = 0; k < 4; ++k) {
        const int u = u0 + k;
        if (val[k]) {
          const float a = alpha[k] + lpb[(b * Tc + (t - 1)) * U1 + u];
          z[k] = a - E[k];
        } else {
          z[k] = -INFINITY;
        }
      }
      float linc[4];
      linc[0] = z[0];
      linc[1] = laef(linc[0], z[1]);
      linc[2] = laef(linc[1], z[2]);
      linc[3] = laef(linc[2], z[3]);
      float lt = linc[3];
#pragma unroll
      for (int off = 1; off < 32; off <<= 1) {
        float n = __shfl_up(lt, off);
        if (lane >= off) lt = laef(lt, n);
      }
      float lex = __shfl_up(lt, 1);
      if (lane == 0) lex = -INFINITY;
#pragma unroll
      for (int k = 0; k < 4; ++k)
        alpha[k] = val[k] ? (E[k] + laef(linc[k], lex)) : -INFINITY;
    }

    if (t == tl) {
#pragma unroll
      for (int k = 0; k < 4; ++k) {
        const int u = u0 + k;
        if (u < U1 && u == ul)
          out[b] = -(alpha[k] + lpb[(b * Tc + t) * U1 + u]);
      }
    }
  }
}

// ---------------------------------------------------------------------------
extern "C" void kernel_launch(void* const* d_in, const int* in_sizes, int n_in,
                              void* d_out, int out_size, void* d_ws, size_t ws_size,
                              hipStream_t stream) {
  (void)in_sizes; (void)n_in; (void)out_size; (void)ws_size;
  const float* logits = (const float*)d_in[0];
  const int* targets  = (const int*)d_in[1];
  const int* loglen   = (const int*)d_in[2];
  const int* tlen     = (const int*)d_in[3];
  float* out = (float*)d_out;

  float* lpb = (float*)d_ws;                       // B*T*(U+1) floats
  float* lpe = lpb + (size_t)Bc * Tc * U1;         // B*T*U floats

  rnnt_lse_kernel<<<1024, 256, 0, stream>>>(logits, targets, lpb, lpe);
  rnnt_dp_kernel<<<Bc, 32, 0, stream>>>(lpb, lpe, loglen, tlen, out);
}